// ResidualGCNLayer_34007551050429
// MI455X (gfx1250) — compile-verified
//
#include <hip/hip_runtime.h>
#include <hip/hip_bf16.h>
#include <math.h>

#define D_DIM 128
#define BN_EPS 1e-5f

typedef __attribute__((ext_vector_type(2))) float v2f;
typedef __attribute__((ext_vector_type(8))) float v8f;

// ---------------------------------------------------------------------------
// 1) init: deg = 1 (self loop), zero BN accumulators
// ---------------------------------------------------------------------------
__global__ void k_init(float* __restrict__ deg, float* __restrict__ sum,
                       float* __restrict__ sumsq, int N) {
  int i = blockIdx.x * blockDim.x + threadIdx.x;
  if (i < N) deg[i] = 1.0f;
  if (i < D_DIM) { sum[i] = 0.0f; sumsq[i] = 0.0f; }
}

// ---------------------------------------------------------------------------
// 2) degree: deg[col[e]] += 1
// ---------------------------------------------------------------------------
__global__ void k_degree(const int* __restrict__ col, float* __restrict__ deg, int E) {
  int e = blockIdx.x * blockDim.x + threadIdx.x;
  if (e < E) atomicAdd(&deg[col[e]], 1.0f);
}

// ---------------------------------------------------------------------------
// 3) deg -> rsqrt(deg) in place (deg >= 1 always, so no zero guard needed)
// ---------------------------------------------------------------------------
__global__ void k_rsqrt(float* __restrict__ deg, int N) {
  int i = blockIdx.x * blockDim.x + threadIdx.x;
  if (i < N) deg[i] = rsqrtf(deg[i]);
}

// ---------------------------------------------------------------------------
// 4) WMMA fp32 GEMM + fused self-loop/bias epilogue:
//      H[N,128]   = X[N,128] @ W[128,128]
//      out[N,128] = bias + H * dinv^2          (self-loop term of GCN aggregate)
//    One wave32 per 16x16 output tile, V_WMMA_F32_16X16X4_F32, 32 K-steps.
//    A (16x4): lanes 0-15 hold M=0..15 {K=0,K=1}, lanes 16-31 hold {K=2,K=3}
//    B (4x16): VGPR0 = rows K=0 (lanes 0-15) / K=2 (lanes 16-31), VGPR1 = K=1/K=3
//    C (16x16): VGPR r -> row r + 8*(lane/16), col lane%16
// ---------------------------------------------------------------------------
__global__ void k_gemm_wmma(const float* __restrict__ X, const float* __restrict__ W,
                            const float* __restrict__ dinv, const float* __restrict__ bias,
                            float* __restrict__ H, float* __restrict__ out, int Nrows) {
  const int wavesPerBlock = blockDim.x >> 5;
  const int wave = blockIdx.x * wavesPerBlock + (threadIdx.x >> 5);
  const int lane = threadIdx.x & 31;
  const int tilesN = D_DIM / 16;                 // 8
  const int tileM = wave / tilesN;
  const int tileN = wave % tilesN;
  if (tileM * 16 >= Nrows) return;               // uniform per wave
  const int m0 = tileM * 16;
  const int n0 = tileN * 16;
  const int half = lane >> 4;                    // 0 or 1
  const int mr = lane & 15;                      // row (A) / col (B,C) within tile

  const float* __restrict__ xrow = X + (size_t)(m0 + mr) * D_DIM;
  v8f c = {};
#pragma unroll
  for (int k0 = 0; k0 < D_DIM; k0 += 4) {
    v2f a, b;
    a[0] = xrow[k0 + 2 * half + 0];
    a[1] = xrow[k0 + 2 * half + 1];
    b[0] = W[(size_t)(k0 + 2 * half + 0) * D_DIM + n0 + mr];
    b[1] = W[(size_t)(k0 + 2 * half + 1) * D_DIM + n0 + mr];
    c = __builtin_amdgcn_wmma_f32_16x16x4_f32(
        /*neg_a=*/false, a, /*neg_b=*/false, b,
        /*c_mod=*/(short)0, c, /*reuse_a=*/false, /*reuse_b=*/false);
  }

  // Fused epilogue: store H, and out = bias + H * dinv[row]^2
  const float bval = bias[n0 + mr];
  float* __restrict__ hout = H + (size_t)m0 * D_DIM + n0;
  float* __restrict__ oout = out + (size_t)m0 * D_DIM + n0;
#pragma unroll
  for (int r = 0; r < 8; ++r) {
    const int rr = r + 8 * half;                 // row within tile
    const float di = dinv[m0 + rr];
    const float hv = c[r];
    hout[(size_t)rr * D_DIM + mr] = hv;
    oout[(size_t)rr * D_DIM + mr] = bval + hv * di * di;
  }
}

// ---------------------------------------------------------------------------
// 5) edge scatter: out[col] += h[row] * dinv[row]*dinv[col]
//    32 lanes per edge, 4 floats per lane (float4 gather, 4 atomic adds)
// ---------------------------------------------------------------------------
__global__ void k_scatter(const int* __restrict__ row, const int* __restrict__ col,
                          const float* __restrict__ H, const float* __restrict__ dinv,
                          float* __restrict__ out, int E) {
  long long idx = (long long)blockIdx.x * blockDim.x + threadIdx.x;
  if (idx >= (long long)E * 32) return;
  int e = (int)(idx >> 5);
  int lane = (int)(idx & 31);
  int r = row[e];
  int cl = col[e];
  float norm = dinv[r] * dinv[cl];
  const float4 hv = *(const float4*)(H + (size_t)r * D_DIM + lane * 4);
  float* op = out + (size_t)cl * D_DIM + lane * 4;
  atomicAdd(op + 0, hv.x * norm);
  atomicAdd(op + 1, hv.y * norm);
  atomicAdd(op + 2, hv.z * norm);
  atomicAdd(op + 3, hv.w * norm);
}

// ---------------------------------------------------------------------------
// 6) BN stats: per-column sum / sumsq (128 threads = 128 columns, coalesced)
// ---------------------------------------------------------------------------
__global__ void k_stats(const float* __restrict__ out, float* __restrict__ sum,
                        float* __restrict__ sumsq, int N, int rowsPerBlock) {
  int d = threadIdx.x;  // 0..127
  int r0 = blockIdx.x * rowsPerBlock;
  int r1 = r0 + rowsPerBlock;
  if (r1 > N) r1 = N;
  float s = 0.0f, sq = 0.0f;
  for (int r = r0; r < r1; ++r) {
    float v = out[(size_t)r * D_DIM + d];
    s += v;
    sq += v * v;
  }
  atomicAdd(&sum[d], s);
  atomicAdd(&sumsq[d], sq);
}

// ---------------------------------------------------------------------------
// 7) finalize: BN(train-mode batch stats) + ReLU + residual, in place on d_out
// ---------------------------------------------------------------------------
__global__ void k_finalize(float* __restrict__ out, const float* __restrict__ x,
                           const float* __restrict__ sum, const float* __restrict__ sumsq,
                           const float* __restrict__ gamma, const float* __restrict__ beta,
                           int N) {
  long long idx = (long long)blockIdx.x * blockDim.x + threadIdx.x;
  if (idx >= (long long)N * D_DIM) return;
  int d = (int)(idx & (D_DIM - 1));
  float invN = 1.0f / (float)N;
  float mean = sum[d] * invN;
  float var = sumsq[d] * invN - mean * mean;
  float v = (out[idx] - mean) * rsqrtf(var + BN_EPS) * gamma[d] + beta[d];
  out[idx] = fmaxf(v, 0.0f) + x[idx];
}

// ---------------------------------------------------------------------------
// launch
// ---------------------------------------------------------------------------
extern "C" void kernel_launch(void* const* d_in, const int* in_sizes, int n_in,
                              void* d_out, int out_size, void* d_ws, size_t ws_size,
                              hipStream_t stream) {
  const float* x     = (const float*)d_in[0];
  const int*   eidx  = (const int*)d_in[1];   // [2, E] flat: row ptr then col ptr
  const float* W     = (const float*)d_in[2];
  const float* bias  = (const float*)d_in[3];
  const float* gamma = (const float*)d_in[4];
  const float* beta  = (const float*)d_in[5];
  float* out = (float*)d_out;

  const int N = in_sizes[0] / D_DIM;   // 50000
  const int E = in_sizes[1] / 2;       // 800000
  const int* row = eidx;
  const int* col = eidx + E;

  // workspace layout (floats): [deg/dinv: N][sum: 128][sumsq: 128][H: N*128]
  float* ws    = (float*)d_ws;
  float* deg   = ws;
  float* sum   = ws + N;
  float* sumsq = ws + N + D_DIM;
  float* H     = ws + N + 2 * D_DIM;

  // 1) init
  k_init<<<(N + 255) / 256, 256, 0, stream>>>(deg, sum, sumsq, N);
  // 2) degree
  k_degree<<<(E + 255) / 256, 256, 0, stream>>>(col, deg, E);
  // 3) dinv = rsqrt(deg)
  k_rsqrt<<<(N + 255) / 256, 256, 0, stream>>>(deg, N);
  // 4) H = x @ W (WMMA fp32) fused with out = bias + H*dinv^2
  {
    int tilesM = (N + 15) / 16;            // 3125 (exact)
    int waves = tilesM * (D_DIM / 16);     // 25000
    int wavesPerBlock = 8;                 // 256 threads
    int blocks = (waves + wavesPerBlock - 1) / wavesPerBlock;
    k_gemm_wmma<<<blocks, wavesPerBlock * 32, 0, stream>>>(x, W, deg, bias, H, out, N);
  }
  // 5) edge scatter-add
  {
    long long total = (long long)E * 32;
    int blocks = (int)((total + 255) / 256);
    k_scatter<<<blocks, 256, 0, stream>>>(row, col, H, deg, out, E);
  }
  // 6) BN column stats (high block count for bandwidth)
  {
    int rowsPerBlock = 32;
    int blocks = (N + rowsPerBlock - 1) / rowsPerBlock;   // 1563
    k_stats<<<blocks, D_DIM, 0, stream>>>(out, sum, sumsq, N, rowsPerBlock);
  }
  // 7) BN + ReLU + residual
  {
    long long total = (long long)N * D_DIM;
    int blocks = (int)((total + 255) / 256);
    k_finalize<<<blocks, 256, 0, stream>>>(out, x, sum, sumsq, gamma, beta, N);
  }
}